// MeanStdMax_45054206935451
// MI455X (gfx1250) — compile-verified
//
#include <hip/hip_runtime.h>
#include <stdint.h>
#include <math.h>

// ---- problem geometry (fixed by the reference) ----
#define BATCH   16
#define LAYERS  13
#define NSLICE  (BATCH * LAYERS)   // 208 independent (b,l) slices
#define SEQ     512                // reduction axis
#define HID     768                // channels per slice
#define OUTSTR  (3 * HID)          // per-slice output: [sum | std | max]

// ---- tiling ----
#define THREADS 256                // 8 waves (wave32)
#define ROWS    16                 // seq rows per tile
#define NTILES  (SEQ / ROWS)       // 32
// bytes per tile = ROWS*HID*4 = 49152; b128 chunks = 3072; per-thread = 12
#define CPT     12
static_assert(ROWS * HID * 4 == CPT * 16 * THREADS, "chunk math");

// CDNA5 async global->LDS path (gfx1250): probe builtins at compile time.
#if defined(__has_builtin)
#if __has_builtin(__builtin_amdgcn_global_load_async_to_lds_b128) && \
    __has_builtin(__builtin_amdgcn_s_wait_asynccnt)
#define USE_ASYNC_LDS 1
#endif
#endif

#ifdef USE_ASYNC_LDS
// Builtin signature (from hipcc diagnostic): arg0 = global v4i* (AS1, non-const),
// arg1 = LDS v4i* (AS3), arg2 = imm offset, arg3 = imm cpol.
typedef int v4i __attribute__((vector_size(16)));
typedef __attribute__((address_space(1))) v4i GV4;   // global
typedef __attribute__((address_space(3))) v4i LV4;   // LDS
#endif

__global__ __launch_bounds__(THREADS)
void MeanStdMax_kernel(const float* __restrict__ in, float* __restrict__ out) {
    const int slice = blockIdx.x;                       // 0..207
    const int tid   = threadIdx.x;
    const float* __restrict__ src = in + (size_t)slice * SEQ * HID;

    // per-thread channels: tid, tid+256, tid+512 (bank-conflict-free in LDS)
    const int c0 = tid, c1 = tid + 256, c2 = tid + 512;

    float s0 = 0.f, s1 = 0.f, s2 = 0.f;                 // sums
    float q0 = 0.f, q1 = 0.f, q2 = 0.f;                 // sums of squares
    float m0 = -__builtin_huge_valf();
    float m1 = -__builtin_huge_valf();
    float m2 = -__builtin_huge_valf();

#ifdef USE_ASYNC_LDS
    __shared__ float smem[2][ROWS][HID];                // 96 KB, double buffer

    // Issue one tile as CPT x b128 async copies per thread (ASYNCcnt += CPT/wave).
    auto issue = [&](int tile, int buf) {
        const float* g = src + (size_t)tile * ROWS * HID;
        float* l = &smem[buf][0][0];
#pragma unroll
        for (int i = 0; i < CPT; ++i) {
            const int chunk = i * THREADS + tid;        // 16-byte chunk id
            __builtin_amdgcn_global_load_async_to_lds_b128(
                (GV4*)(g + chunk * 4), (LV4*)(l + chunk * 4),
                /*offset=*/0, /*cpol=*/0);
        }
    };

    issue(0, 0);
    issue(1, 1);

    for (int t = 0; t < NTILES; ++t) {
        const int cur = t & 1;
        // Wait until only the *next* tile's copies are still outstanding
        // (async loads complete in order within a wave).
        if (t + 1 < NTILES) __builtin_amdgcn_s_wait_asynccnt(CPT);
        else                __builtin_amdgcn_s_wait_asynccnt(0);
        __syncthreads();                                // tile t visible to all waves

#pragma unroll
        for (int s = 0; s < ROWS; ++s) {
            const float a0 = smem[cur][s][c0];
            const float a1 = smem[cur][s][c1];
            const float a2 = smem[cur][s][c2];
            s0 += a0; q0 = fmaf(a0, a0, q0); m0 = fmaxf(m0, a0);
            s1 += a1; q1 = fmaf(a1, a1, q1); m1 = fmaxf(m1, a1);
            s2 += a2; q2 = fmaf(a2, a2, q2); m2 = fmaxf(m2, a2);
        }

        __syncthreads();                                // all waves done reading buf
        if (t + 2 < NTILES) issue(t + 2, cur);          // refill this buffer
    }
#else
    // Fallback: direct coalesced global loads (still bandwidth-optimal).
    for (int s = 0; s < SEQ; ++s) {
        const float* __restrict__ r = src + (size_t)s * HID;
        const float a0 = r[c0], a1 = r[c1], a2 = r[c2];
        s0 += a0; q0 = fmaf(a0, a0, q0); m0 = fmaxf(m0, a0);
        s1 += a1; q1 = fmaf(a1, a1, q1); m1 = fmaxf(m1, a1);
        s2 += a2; q2 = fmaf(a2, a2, q2); m2 = fmaxf(m2, a2);
    }
#endif

    // std (unbiased, ddof=1): sqrt((sum_sq - sum^2/N) / (N-1))
    const float invN = 1.0f / (float)SEQ;
    const float invNm1 = 1.0f / (float)(SEQ - 1);
    const float v0 = fmaxf((q0 - s0 * s0 * invN) * invNm1, 0.f);
    const float v1 = fmaxf((q1 - s1 * s1 * invN) * invNm1, 0.f);
    const float v2 = fmaxf((q2 - s2 * s2 * invN) * invNm1, 0.f);

    float* __restrict__ o = out + (size_t)slice * OUTSTR;
    o[c0]            = s0;  o[c1]            = s1;  o[c2]            = s2;
    o[HID + c0]      = sqrtf(v0); o[HID + c1] = sqrtf(v1); o[HID + c2] = sqrtf(v2);
    o[2 * HID + c0]  = m0;  o[2 * HID + c1]  = m1;  o[2 * HID + c2]  = m2;
}

extern "C" void kernel_launch(void* const* d_in, const int* in_sizes, int n_in,
                              void* d_out, int out_size, void* d_ws, size_t ws_size,
                              hipStream_t stream) {
    (void)in_sizes; (void)n_in; (void)out_size; (void)d_ws; (void)ws_size;
    const float* in = (const float*)d_in[0];
    float* out = (float*)d_out;
    MeanStdMax_kernel<<<dim3(NSLICE), dim3(THREADS), 0, stream>>>(in, out);
}